// ResNet_PCD_30099130810370
// MI455X (gfx1250) — compile-verified
//
#include <hip/hip_runtime.h>
#include <hip/hip_bf16.h>

// ResNet_PCD: 1-NN (131072 points -> 2048 nodes, 3D) + stable top-32 grouping.
// Core math mapped onto V_WMMA_F32_16X16X4_F32 (CDNA5, wave32):
//   A row (point)  = (x, y, z, 1)
//   B col (node)   = (-2nx, -2ny, -2nz, |n|^2)
//   D = A*B        = -2 p.n + |n|^2   (add |p|^2 at the end for min_d2)
// Each wave owns 4 point-tiles (64 points) so every LDS B-fragment load feeds
// 4 independent WMMAs; B is double-buffered to hide ds latency.

#define N_POINTS 131072
#define N_NODES  2048
#define KSEL     32
#define PT       4            // point tiles (of 16) per wave

typedef __attribute__((ext_vector_type(2))) float v2f;
typedef __attribute__((ext_vector_type(8))) float v8f;

// ---------------------------------------------------------------- kernel 1
__global__ void prep_nodes(const float* __restrict__ nodes,
                           float4* __restrict__ aug) {
    int i = blockIdx.x * blockDim.x + threadIdx.x;
    if (i < N_NODES) {
        float x = nodes[3 * i + 0];
        float y = nodes[3 * i + 1];
        float z = nodes[3 * i + 2];
        aug[i] = make_float4(-2.0f * x, -2.0f * y, -2.0f * z,
                             x * x + y * y + z * z);
    }
}

// ---------------------------------------------------------------- kernel 2
__global__ void __launch_bounds__(256)
knn_wmma(const float* __restrict__ pc,
         const float4* __restrict__ aug,
         float* __restrict__ out_d2,
         float* __restrict__ out_id_f,
         int* __restrict__ out_id_i) {
    __shared__ float4 s_aug[N_NODES];          // 32 KB of 320 KB WGP LDS

    const int tid = threadIdx.x;
    for (int i = tid; i < N_NODES; i += 256) s_aug[i] = aug[i];
    __syncthreads();

    const int lane = tid & 31;
    const int wave = tid >> 5;
    const int half = lane >> 4;   // 0: components 0,1 ; 1: components 2,3
    const int nl   = lane & 15;
    const int ptBase = (blockIdx.x * 8 + wave) * (PT * 16);

    // A fragments (ISA 16x4 f32 layout): lanes 0-15 (x,y), lanes 16-31 (z,1)
    v2f a[PT];
#pragma unroll
    for (int j = 0; j < PT; ++j) {
        const int m = ptBase + j * 16 + nl;
        a[j].x = pc[3 * m + 2 * half];
        a[j].y = half ? 1.0f : pc[3 * m + 1];
    }

    float minv[PT][8];
    int   mini[PT][8];
#pragma unroll
    for (int j = 0; j < PT; ++j)
#pragma unroll
        for (int r = 0; r < 8; ++r) { minv[j][r] = 3.4e38f; mini[j][r] = 0; }

    const float2* s2 = (const float2*)s_aug;

    // double-buffered B fragment: lanes 0-15 (-2nx,-2ny), lanes 16-31 (-2nz,|n|^2)
    float2 bb = s2[(nl << 1) + half];          // tile 0

    for (int t = 0; t < N_NODES / 16; ++t) {
        const int n  = (t << 4) + nl;
        const int tn = (t + 1 < N_NODES / 16) ? t + 1 : t;
        float2 bbn = s2[((((tn << 4) + nl)) << 1) + half];   // prefetch next

        v2f b; b.x = bb.x; b.y = bb.y;
#pragma unroll
        for (int j = 0; j < PT; ++j) {
            v8f c = {};
            c = __builtin_amdgcn_wmma_f32_16x16x4_f32(
                    false, a[j], false, b, (short)0, c, false, false);
            // D: this lane holds rows M = r + 8*half, column N = nl (node n)
#pragma unroll
            for (int r = 0; r < 8; ++r) {
                const float v  = c[r];
                const bool  lt = v < minv[j][r];
                minv[j][r] = lt ? v : minv[j][r];
                mini[j][r] = lt ? n : mini[j][r];
            }
        }
        bb = bbn;
    }

    // Branchless butterfly min+argmin across the 16 columns of each half-wave
    // (tie -> lower node id, matching argmin-first semantics).
#pragma unroll
    for (int off = 1; off < 16; off <<= 1) {
#pragma unroll
        for (int j = 0; j < PT; ++j)
#pragma unroll
            for (int r = 0; r < 8; ++r) {
                const float ov = __shfl_xor(minv[j][r], off, 32);
                const int   oi = __shfl_xor(mini[j][r], off, 32);
                const bool take = (ov < minv[j][r]) |
                                  ((ov == minv[j][r]) & (oi < mini[j][r]));
                minv[j][r] = take ? ov : minv[j][r];
                mini[j][r] = take ? oi : mini[j][r];
            }
    }

    if (nl == 0) {   // lane 0 -> rows 0-7 of each tile, lane 16 -> rows 8-15
#pragma unroll
        for (int j = 0; j < PT; ++j)
#pragma unroll
            for (int r = 0; r < 8; ++r) {
                const int p = ptBase + j * 16 + (half << 3) + r;
                const float x = pc[3 * p + 0];
                const float y = pc[3 * p + 1];
                const float z = pc[3 * p + 2];
                out_d2[p]   = minv[j][r] + x * x + y * y + z * z;
                out_id_f[p] = (float)mini[j][r];
                out_id_i[p] = mini[j][r];
            }
    }
}

// ---------------------------------------------------------------- kernel 3
// One wave per node. Ballot + prefix popcount over points in ascending order
// reproduces the stable-sort grouping; slots beyond count stay 0.
__global__ void __launch_bounds__(256)
group_patch(const int* __restrict__ pcd_id,
            float* __restrict__ patch_out) {
    const int lane = threadIdx.x & 31;
    const int node = (blockIdx.x * blockDim.x + threadIdx.x) >> 5;
    if (node >= N_NODES) return;

    patch_out[node * KSEL + lane] = 0.0f;  // same-wave stores stay ordered

    int count = 0;
    for (int base = 0; base < N_POINTS; base += 32) {
        const int id = pcd_id[base + lane];
        const bool match = (id == node);
        const unsigned mask = (unsigned)__ballot(match);
        if (match) {
            const int pos = count + __popc(mask & ((1u << lane) - 1u));
            if (pos < KSEL) patch_out[node * KSEL + pos] = (float)(base + lane);
        }
        count += __popc(mask);
        if (count >= KSEL) break;
    }
}

// ---------------------------------------------------------------- launch
extern "C" void kernel_launch(void* const* d_in, const int* in_sizes, int n_in,
                              void* d_out, int out_size, void* d_ws, size_t ws_size,
                              hipStream_t stream) {
    const float* pc    = (const float*)d_in[0];  // (131072, 3) f32
    const float* nodes = (const float*)d_in[1];  // (2048, 3)  f32
    (void)in_sizes; (void)n_in; (void)out_size; (void)ws_size;

    float* out     = (float*)d_out;
    float* out_d2  = out;                  // 131072
    float* out_id  = out + N_POINTS;       // 131072 (indices as float)
    float* out_pat = out + 2 * N_POINTS;   // 2048*32 (indices as float)

    float4* aug   = (float4*)d_ws;                                    // 32 KB
    int*    pcd_i = (int*)((char*)d_ws + N_NODES * sizeof(float4));   // 512 KB

    prep_nodes<<<(N_NODES + 255) / 256, 256, 0, stream>>>(nodes, aug);
    knn_wmma<<<N_POINTS / (16 * PT) / 8, 256, 0, stream>>>(pc, aug, out_d2, out_id, pcd_i);
    group_patch<<<(N_NODES * 32 + 255) / 256, 256, 0, stream>>>(pcd_i, out_pat);
}